// Encoder_85349590106290
// MI455X (gfx1250) — compile-verified
//
#include <hip/hip_runtime.h>
#include <hip/hip_bf16.h>

#define NN 100000
#define NE 1200000
#define DIM 64
#define NLAYERS 3
#define OUTD (NLAYERS * DIM)
#define BN_EPS 1e-5f

typedef __attribute__((ext_vector_type(2))) float v2f;
typedef __attribute__((ext_vector_type(4))) float v4f;
typedef __attribute__((ext_vector_type(8))) float v8f;

// ---------------------------------------------------------------------------
// agg <- h (float4), zero BN stats accumulator (128 floats: sum | sumsq)
// ---------------------------------------------------------------------------
__global__ void gin_init_kernel(const float* __restrict__ h_in,
                                float* __restrict__ agg,
                                float* __restrict__ gstats) {
    size_t i = (size_t)blockIdx.x * blockDim.x + threadIdx.x;
    if (i < (size_t)NN * DIM / 4) {
        ((v4f*)agg)[i] = ((const v4f*)h_in)[i];
    }
    if (i < 128) gstats[i] = 0.0f;
}

// ---------------------------------------------------------------------------
// agg[dst] += h[src]  — one thread per (edge, feature); fp32 HW atomics.
// ---------------------------------------------------------------------------
__global__ void gin_scatter_kernel(const float* __restrict__ h_in,
                                   const int* __restrict__ ei,
                                   float* __restrict__ agg) {
    long long tid = (long long)blockIdx.x * blockDim.x + threadIdx.x;
    if (tid >= (long long)NE * DIM) return;
    int e = (int)(tid >> 6);
    int f = (int)(tid & 63);
    int s = ei[e];
    int d = ei[NE + e];
    unsafeAtomicAdd(&agg[(size_t)d * DIM + f], h_in[(size_t)s * DIM + f]);
}

// ---------------------------------------------------------------------------
// Fused MLP: z = relu( relu(agg@W1 + b1) @ W2 + b2 ), plus BN sum/sumsq.
// 256 threads = 8 waves; each wave computes a 16x64 output tile with
// v_wmma_f32_16x16x4_f32 (16 K-steps per GEMM, 4 N-tiles of 16).
//
// Weights are staged into LDS K-pair-interleaved:
//   Wp[m*128 + col*2 + p] = W[(2m+p)*64 + col]   (m = k>>1, p = k&1)
// so every B fragment {W[ks][col], W[ks+1][col]} is one aligned ds_load_b64
// landing in an even VGPR pair — no operand re-packing movs before WMMA.
//
// LDS overlay: regA holds packed W1 during GEMM1, then relu(z1) staging.
// A-feed is branch-free (clamped pointer + 0/1 mask multiply).
// ---------------------------------------------------------------------------
__global__ void gin_mlp_kernel(const float* __restrict__ agg,
                               const float* __restrict__ W1,
                               const float* __restrict__ b1,
                               const float* __restrict__ W2,
                               const float* __restrict__ b2,
                               float* __restrict__ z,
                               float* __restrict__ gstats,
                               int n) {
    __shared__ float W2s[DIM * DIM];     // 16 KB, packed interleaved
    __shared__ float regA[8 * 16 * DIM]; // 32 KB: packed W1, then z1 tiles
    __shared__ float b1s[DIM], b2s[DIM];
    __shared__ float sstats[128];

    const int t = threadIdx.x;
    // stage weights with K-pair interleaving (float4 global loads)
    {
        const v4f* w1v = (const v4f*)W1;
        const v4f* w2v = (const v4f*)W2;
        #pragma unroll
        for (int i = 0; i < 4; ++i) {
            int idx = t + 256 * i;        // float4 index 0..1023
            int k   = idx >> 4;           // row of W (0..63)
            int col = (idx << 2) & 63;    // first of 4 consecutive cols
            int m = k >> 1, p = k & 1;
            v4f w1 = w1v[idx];
            v4f w2 = w2v[idx];
            #pragma unroll
            for (int q = 0; q < 4; ++q) {
                regA[m * 128 + (col + q) * 2 + p] = w1[q];
                W2s [m * 128 + (col + q) * 2 + p] = w2[q];
            }
        }
    }
    if (t < DIM) { b1s[t] = b1[t]; b2s[t] = b2[t]; }
    if (t < 128) sstats[t] = 0.0f;
    __syncthreads();

    const int wave = t >> 5;
    const int lane = t & 31;
    const int half = lane >> 4;   // selects K-pair / row-group per ISA layout
    const int l15  = lane & 15;
    const int r0   = (blockIdx.x * 8 + wave) * 16;
    const bool full = (r0 + 16 <= n);   // wave-uniform

    // ---------------- GEMM1: relu(agg @ W1 + b1) ----------------
    v8f acc[4];
    #pragma unroll
    for (int nt = 0; nt < 4; ++nt) {
        float bv = b1s[nt * 16 + l15];
        #pragma unroll
        for (int j = 0; j < 8; ++j) acc[nt][j] = bv;
    }

    const int arow = r0 + l15;
    const float amask = (arow < n) ? 1.0f : 0.0f;
    const int arow_c = (arow < n) ? arow : (n - 1);
    const float* aptr = agg + (size_t)arow_c * DIM;

    #pragma unroll
    for (int kk = 0; kk < 16; ++kk) {
        const int m = kk * 2 + half;                 // K-pair index
        v2f a = *(const v2f*)(aptr + kk * 4 + half * 2);  // b64 load
        a.x *= amask;
        a.y *= amask;
        #pragma unroll
        for (int nt = 0; nt < 4; ++nt) {
            v2f b = *(const v2f*)(&regA[m * 128 + (nt * 16 + l15) * 2]);
            acc[nt] = __builtin_amdgcn_wmma_f32_16x16x4_f32(
                false, a, false, b, (short)0, acc[nt], false, false);
        }
    }

    __syncthreads();  // all waves done reading packed W1 from regA

    // stage relu(z1) tile for this wave into regA (row-major 16x64)
    float* zw = &regA[wave * 16 * DIM];
    #pragma unroll
    for (int nt = 0; nt < 4; ++nt)
        #pragma unroll
        for (int j = 0; j < 8; ++j) {
            float v = acc[nt][j];
            zw[(j + 8 * half) * DIM + nt * 16 + l15] = v > 0.0f ? v : 0.0f;
        }
    __syncthreads();

    // ---------------- GEMM2: relu(z1 @ W2 + b2) ----------------
    #pragma unroll
    for (int nt = 0; nt < 4; ++nt) {
        float bv = b2s[nt * 16 + l15];
        #pragma unroll
        for (int j = 0; j < 8; ++j) acc[nt][j] = bv;
    }
    #pragma unroll
    for (int kk = 0; kk < 16; ++kk) {
        const int m = kk * 2 + half;
        v2f a = *(const v2f*)(zw + l15 * DIM + kk * 4 + half * 2);  // b64
        #pragma unroll
        for (int nt = 0; nt < 4; ++nt) {
            v2f b = *(const v2f*)(&W2s[m * 128 + (nt * 16 + l15) * 2]);
            acc[nt] = __builtin_amdgcn_wmma_f32_16x16x4_f32(
                false, a, false, b, (short)0, acc[nt], false, false);
        }
    }

    // relu, store z, per-channel partial stats (fast path: branch-free)
    if (full) {
        #pragma unroll
        for (int nt = 0; nt < 4; ++nt) {
            float s = 0.0f, s2 = 0.0f;
            #pragma unroll
            for (int j = 0; j < 8; ++j) {
                float v = acc[nt][j];
                v = v > 0.0f ? v : 0.0f;
                int row = r0 + j + 8 * half;
                z[(size_t)row * DIM + nt * 16 + l15] = v;
                s  += v;
                s2 += v * v;
            }
            unsafeAtomicAdd(&sstats[nt * 16 + l15], s);
            unsafeAtomicAdd(&sstats[64 + nt * 16 + l15], s2);
        }
    } else {
        #pragma unroll
        for (int nt = 0; nt < 4; ++nt) {
            float s = 0.0f, s2 = 0.0f;
            #pragma unroll
            for (int j = 0; j < 8; ++j) {
                float v = acc[nt][j];
                v = v > 0.0f ? v : 0.0f;
                int row = r0 + j + 8 * half;
                if (row < n) {
                    z[(size_t)row * DIM + nt * 16 + l15] = v;
                    s  += v;
                    s2 += v * v;
                }
            }
            unsafeAtomicAdd(&sstats[nt * 16 + l15], s);
            unsafeAtomicAdd(&sstats[64 + nt * 16 + l15], s2);
        }
    }
    __syncthreads();
    if (t < 128) unsafeAtomicAdd(&gstats[t], sstats[t]);
}

// ---------------------------------------------------------------------------
// BatchNorm (training stats, biased var) + write h (next layer) and out slice
// float4 per thread: 4 consecutive channels of one row.
// ---------------------------------------------------------------------------
__global__ void gin_bn_kernel(const float* __restrict__ z,
                              const float* __restrict__ gstats,
                              const float* __restrict__ gamma,
                              const float* __restrict__ beta,
                              float* __restrict__ h,
                              float* __restrict__ out,
                              int layer) {
    size_t tid = (size_t)blockIdx.x * blockDim.x + threadIdx.x;
    if (tid >= (size_t)NN * DIM / 4) return;
    size_t i4 = tid * 4;
    int c = (int)(i4 & 63);
    size_t r = i4 >> 6;
    const float invN = 1.0f / (float)NN;

    v4f zv = *(const v4f*)(z + i4);
    v4f o;
    #pragma unroll
    for (int k = 0; k < 4; ++k) {
        float mean = gstats[c + k] * invN;
        float var  = gstats[64 + c + k] * invN - mean * mean;
        float inv  = rsqrtf(var + BN_EPS);
        o[k] = (zv[k] - mean) * inv * gamma[c + k] + beta[c + k];
    }
    *(v4f*)(h + i4) = o;
    *(v4f*)(out + r * OUTD + (size_t)layer * DIM + c) = o;
}

// ---------------------------------------------------------------------------
extern "C" void kernel_launch(void* const* d_in, const int* in_sizes, int n_in,
                              void* d_out, int out_size, void* d_ws, size_t ws_size,
                              hipStream_t stream) {
    const float* x     = (const float*)d_in[0];
    const int*   ei    = (const int*)d_in[1];
    // d_in[2] = batch (unused by the reference computation)
    const float* W1    = (const float*)d_in[3];
    const float* b1    = (const float*)d_in[4];
    const float* W2    = (const float*)d_in[5];
    const float* b2    = (const float*)d_in[6];
    const float* gamma = (const float*)d_in[7];
    const float* beta  = (const float*)d_in[8];
    float* out = (float*)d_out;

    float* ws  = (float*)d_ws;
    float* h      = ws;                                  // [NN, 64]
    float* agg    = h   + (size_t)NN * DIM;              // [NN, 64]
    float* z      = agg + (size_t)NN * DIM;              // [NN, 64]
    float* gstats = z   + (size_t)NN * DIM;              // [128]

    const int vec4 = NN * DIM / 4;                       // 1.6M
    const dim3 blk(256);
    const dim3 grid_vec4((vec4 + 255) / 256);
    const dim3 grid_edge((unsigned)(((long long)NE * DIM + 255) / 256));
    const dim3 grid_mlp((NN + 127) / 128);               // 128 rows per block

    for (int layer = 0; layer < NLAYERS; ++layer) {
        const float* h_in = (layer == 0) ? x : h;
        gin_init_kernel<<<grid_vec4, blk, 0, stream>>>(h_in, agg, gstats);
        gin_scatter_kernel<<<grid_edge, blk, 0, stream>>>(h_in, ei, agg);
        gin_mlp_kernel<<<grid_mlp, blk, 0, stream>>>(
            agg,
            W1 + (size_t)layer * DIM * DIM, b1 + (size_t)layer * DIM,
            W2 + (size_t)layer * DIM * DIM, b2 + (size_t)layer * DIM,
            z, gstats, NN);
        gin_bn_kernel<<<grid_vec4, blk, 0, stream>>>(
            z, gstats, gamma + (size_t)layer * DIM, beta + (size_t)layer * DIM,
            h, out, layer);
    }
}